// RBFDistanceAttention_25099788877990
// MI455X (gfx1250) — compile-verified
//
#include <hip/hip_runtime.h>
#include <math.h>

typedef __attribute__((ext_vector_type(2))) float v2f;
typedef __attribute__((ext_vector_type(4))) float v4f;
typedef __attribute__((ext_vector_type(8))) float v8f;
typedef __attribute__((ext_vector_type(4))) int   v4i;

#define BB      4
#define LL      4096
#define DD      128
#define QB      16          // query rows per workgroup
#define NKT     (LL / 16)   // 256 key tiles
#define NWAVES  8
#define THREADS 256

// LDS (floats): sS[QB][LL] | kn2[LL] | sQ[QB][DD] | sQn[QB] | red[QB][16] | rowmax[QB] | rowinv[QB]
#define SMEM_FLOATS (QB*LL + LL + QB*DD + QB + QB*16 + QB + QB)

#if __has_builtin(__builtin_amdgcn_global_load_async_to_lds_b128) && \
    __has_builtin(__builtin_amdgcn_s_wait_asynccnt)
#define HAVE_ASYNC_LDS 1
typedef __attribute__((address_space(1))) v4i* as1_v4i_p;
typedef __attribute__((address_space(3))) v4i* as3_v4i_p;
#endif

__global__ __launch_bounds__(THREADS, 1)
void rbf_attn_kernel(const float* __restrict__ Q, const float* __restrict__ K,
                     const float* __restrict__ V, const int* __restrict__ mask,
                     float* __restrict__ out, float* __restrict__ attn,
                     float* __restrict__ scores, float* __restrict__ dist2)
{
    extern __shared__ float smem[];
    float* sS     = smem;                 // [QB][LL] dot strip -> score strip -> attn strip
    float* kn2    = sS  + QB * LL;        // [LL]
    float* sQ     = kn2 + LL;             // [QB][DD]
    float* sQn    = sQ  + QB * DD;        // [QB]
    float* red    = sQn + QB;             // [QB][16]
    float* rowmax = red + QB * 16;        // [QB]
    float* rowinv = rowmax + QB;          // [QB]

    const int b    = blockIdx.y;
    const int q0   = blockIdx.x * QB;
    const int tid  = threadIdx.x;
    const int wave = tid >> 5;
    const int lane = tid & 31;
    const int mrow = lane & 15;           // A: M row / B: N col owned by this lane
    const int ksel = (lane >> 4) * 2;     // K-dim sub-offset (0 or 2)
    const int rbase = (lane >> 4) * 8;    // C/D: M = v + rbase

    const float* Qb = Q + (size_t)b * LL * DD;
    const float* Kb = K + (size_t)b * LL * DD;
    const float* Vb = V + (size_t)b * LL * DD;
    const int*   Mb = mask + (size_t)b * LL * LL;
    float* outb    = out    + (size_t)b * LL * DD;
    float* attnb   = attn   + (size_t)b * LL * LL;
    float* scoresb = scores + (size_t)b * LL * LL;
    float* dist2b  = dist2  + (size_t)b * LL * LL;

    // ---- stage Q block into LDS ----
    {
        const v4f* Qv  = (const v4f*)(Qb + (size_t)q0 * DD);
        v4f*       sQv = (v4f*)sQ;
#ifdef HAVE_ASYNC_LDS
        for (int i = tid; i < QB * DD / 4; i += THREADS) {
            __builtin_amdgcn_global_load_async_to_lds_b128(
                (as1_v4i_p)(Qv + i), (as3_v4i_p)(sQv + i), 0, 0);
        }
        __builtin_amdgcn_s_wait_asynccnt(0);
#else
        for (int i = tid; i < QB * DD / 4; i += THREADS) sQv[i] = Qv[i];
#endif
    }
    // ---- k_norm^2 for all keys of this batch (L2-resident reads) ----
    for (int k = tid; k < LL; k += THREADS) {
        const v4f* kr = (const v4f*)(Kb + (size_t)k * DD);
        float s = 0.f;
        #pragma unroll 4
        for (int d4 = 0; d4 < DD / 4; ++d4) {
            v4f v = kr[d4];
            s += v.x * v.x + v.y * v.y + v.z * v.z + v.w * v.w;
        }
        kn2[k] = s;
    }
    __syncthreads();
    // ---- q_norm^2 for the 16 query rows ----
    if (tid < QB) {
        float s = 0.f;
        for (int d = 0; d < DD; ++d) { float v = sQ[tid * DD + d]; s += v * v; }
        sQn[tid] = s;
    }
    __syncthreads();

    // ---- Phase 1: raw dot = Q K^T via f32 WMMA; park dots in LDS strip ----
    for (int kt = wave; kt < NKT; kt += NWAVES) {
        const int k0 = kt * 16;
        v8f c = {};
        const float* Arow = sQ + mrow * DD + ksel;                 // Q row fragment base
        const float* Brow = Kb + (size_t)(k0 + mrow) * DD + ksel;  // K row fragment base
        // prefetch this lane's K row for the next tile this wave will process
        if (kt + NWAVES < NKT)
            __builtin_prefetch(Brow + (size_t)NWAVES * 16 * DD, 0, 1);
        #pragma unroll 8
        for (int d0 = 0; d0 < DD; d0 += 4) {
            v2f a;  a.x = Arow[d0];  a.y = Arow[d0 + 1];
            v2f bf; bf.x = Brow[d0]; bf.y = Brow[d0 + 1];
            c = __builtin_amdgcn_wmma_f32_16x16x4_f32(false, a, false, bf,
                                                      (short)0, c, false, false);
        }
        const int col = k0 + mrow;
        #pragma unroll
        for (int v = 0; v < 8; ++v)
            sS[(rbase + v) * LL + col] = c[v];
    }
    __syncthreads();

    // ---- Phase 1.5: coalesced vectorized sweep: dist2 / mask / scores ----
    for (int i = tid * 4; i < QB * LL; i += THREADS * 4) {
        const int r   = i >> 12;           // i / LL
        const int col = i & (LL - 1);
        const size_t g = (size_t)(q0 + r) * LL + col;
        v4f   dot = *(const v4f*)(sS + i);
        float qn  = sQn[r];
        v4f   kn  = *(const v4f*)(kn2 + col);
        v4f dd;
        dd.x = qn + kn.x - 2.0f * dot.x;
        dd.y = qn + kn.y - 2.0f * dot.y;
        dd.z = qn + kn.z - 2.0f * dot.z;
        dd.w = qn + kn.w - 2.0f * dot.w;
        __builtin_nontemporal_store(dd, (v4f*)(dist2b + g));
        v4i mk = *(const v4i*)(Mb + g);
        const float ninf = -__builtin_inff();
        v4f s;
        s.x = (mk.x == 0) ? ninf : dd.x * (-1.0f / 256.0f);   // -dist2/(2*sigma^2), sigma^2=D
        s.y = (mk.y == 0) ? ninf : dd.y * (-1.0f / 256.0f);
        s.z = (mk.z == 0) ? ninf : dd.z * (-1.0f / 256.0f);
        s.w = (mk.w == 0) ? ninf : dd.w * (-1.0f / 256.0f);
        __builtin_nontemporal_store(s, (v4f*)(scoresb + g));
        *(v4f*)(sS + i) = s;
    }
    __syncthreads();

    // ---- Phase 2: row max / row sum-exp over the LDS strip (v4f scans) ----
    {
        const int r = tid >> 4, ch = tid & 15;
        const v4f* row4 = (const v4f*)(sS + r * LL + ch * (LL / 16));
        float m = -__builtin_inff();
        for (int i = 0; i < LL / 64; ++i) {
            v4f v = row4[i];
            m = fmaxf(fmaxf(fmaxf(m, v.x), v.y), fmaxf(v.z, v.w));
        }
        red[r * 16 + ch] = m;
    }
    __syncthreads();
    if (tid < QB) {
        float m = -__builtin_inff();
        for (int i = 0; i < 16; ++i) m = fmaxf(m, red[tid * 16 + i]);
        rowmax[tid] = m;
    }
    __syncthreads();
    {
        const int r = tid >> 4, ch = tid & 15;
        const float m = rowmax[r];
        const v4f* row4 = (const v4f*)(sS + r * LL + ch * (LL / 16));
        float s = 0.f;
        for (int i = 0; i < LL / 64; ++i) {
            v4f v = row4[i];
            s += __expf(v.x - m) + __expf(v.y - m) + __expf(v.z - m) + __expf(v.w - m);
        }
        red[r * 16 + ch] = s;
    }
    __syncthreads();
    if (tid < QB) {
        float s = 0.f;
        for (int i = 0; i < 16; ++i) s += red[tid * 16 + i];
        rowinv[tid] = 1.0f / s;
    }
    __syncthreads();

    // ---- Phase 3a: coalesced attn write; overwrite LDS strip with attn ----
    for (int i = tid * 4; i < QB * LL; i += THREADS * 4) {
        const int r   = i >> 12;
        const int col = i & (LL - 1);
        const float m = rowmax[r], inv = rowinv[r];
        v4f s = *(const v4f*)(sS + i);
        v4f a;
        a.x = __expf(s.x - m) * inv;
        a.y = __expf(s.y - m) * inv;
        a.z = __expf(s.z - m) * inv;
        a.w = __expf(s.w - m) * inv;
        *(v4f*)(sS + i) = a;
        __builtin_nontemporal_store(a, (v4f*)(attnb + (size_t)(q0 + r) * LL + col));
    }
    __syncthreads();

    // ---- Phase 3b: out = attn @ V via f32 WMMA; one 16-col tile per wave ----
    {
        const int n0 = wave * 16;
        v8f c = {};
        const float* Arow = sS + mrow * LL + ksel;      // attn row fragment base
        const float* Vcol = Vb + n0 + mrow;             // V column owned by this lane
        #pragma unroll 8
        for (int k0 = 0; k0 < LL; k0 += 4) {
            v2f a;  a.x = Arow[k0]; a.y = Arow[k0 + 1];
            v2f bf;
            bf.x = Vcol[(size_t)(k0 + ksel) * DD];
            bf.y = Vcol[(size_t)(k0 + ksel + 1) * DD];
            c = __builtin_amdgcn_wmma_f32_16x16x4_f32(false, a, false, bf,
                                                      (short)0, c, false, false);
        }
        #pragma unroll
        for (int v = 0; v < 8; ++v)
            outb[(size_t)(q0 + rbase + v) * DD + n0 + mrow] = c[v];
    }
}

extern "C" void kernel_launch(void* const* d_in, const int* in_sizes, int n_in,
                              void* d_out, int out_size, void* d_ws, size_t ws_size,
                              hipStream_t stream) {
    const float* Q    = (const float*)d_in[0];
    const float* K    = (const float*)d_in[1];
    const float* V    = (const float*)d_in[2];
    const int*   mask = (const int*)d_in[3];

    float* out    = (float*)d_out;                         // [B,L,D]
    float* attn   = out    + (size_t)BB * LL * DD;         // [B,L,L]
    float* scores = attn   + (size_t)BB * LL * LL;         // [B,L,L]
    float* dist2  = scores + (size_t)BB * LL * LL;         // [B,L,L]

    const size_t shmem = (size_t)SMEM_FLOATS * sizeof(float);   // ~281 KB < 320 KB WGP LDS
    (void)hipFuncSetAttribute((const void*)rbf_attn_kernel,
                              hipFuncAttributeMaxDynamicSharedMemorySize, (int)shmem);

    dim3 grid(LL / QB, BB);   // 256 x 4 workgroups
    rbf_attn_kernel<<<grid, THREADS, shmem, stream>>>(Q, K, V, mask,
                                                      out, attn, scores, dist2);
}